// Loss_14456859918817
// MI455X (gfx1250) — compile-verified
//
#include <hip/hip_runtime.h>

typedef __attribute__((ext_vector_type(2))) float v2f;
typedef __attribute__((ext_vector_type(8))) float v8f;

#define G_DIM   7
#define NB      2
#define CLS     20
#define CH      (5*NB+CLS)      /* 30 */
#define B_IMG   16384
#define TGT_PER 8
#define T_TOT   (B_IMG*TGT_PER) /* 131072 */
#define LAMBDA_COORD 5.0f
#define LAMBDA_NOOBJ 0.5f

// ---------------------------------------------------------------------------
// Kernel 1: per-target loss terms (gather + IoU + argmax over NB=2 boxes).
// One thread per target; deterministic block tree-reduction into partial[].
// ---------------------------------------------------------------------------
__global__ void yolo_target_loss(const float* __restrict__ out,
                                 const float* __restrict__ tgt,
                                 float* __restrict__ partial, int n) {
    __shared__ float sdata[256];
    int t = blockIdx.x * 256 + threadIdx.x;
    float loss = 0.0f;
    if (t < n) {
        const float* tr = tgt + (size_t)t * 8;
        float x_t = tr[0], y_t = tr[1], w_t = tr[2], h_t = tr[3];
        int gx = (int)tr[4], gy = (int)tr[5], cl = (int)tr[6], bid = (int)tr[7];
        const float* cell = out + ((((size_t)bid * G_DIM) + gx) * G_DIM + gy) * CH;

        float top_t  = y_t - 3.5f * h_t, bot_t   = y_t + 3.5f * h_t;
        float left_t = x_t - 3.5f * w_t, right_t = x_t + 3.5f * w_t;
        float area_t = w_t * h_t * 49.0f;

        float best = -1.0f;
        float xr = 0.f, yr = 0.f, wr = 0.f, hr = 0.f, cr = 0.f;
#pragma unroll
        for (int b = 0; b < NB; ++b) {
            float xg = cell[5*b+0], yg = cell[5*b+1];
            float wg = cell[5*b+2], hg = cell[5*b+3], cg = cell[5*b+4];
            float topg  = yg - 3.5f*hg, botg   = yg + 3.5f*hg;
            float leftg = xg - 3.5f*wg, rightg = xg + 3.5f*wg;
            float wi = fmaxf(fminf(rightg, right_t) - fmaxf(leftg, left_t), 0.0f);
            float hi = fmaxf(fmaxf(topg,  top_t)   - fminf(botg,  bot_t),  0.0f);
            float ai = wi * hi;
            float ag = wg * hg * 49.0f;
            float at = area_t + ag - ai;
            float iou = (at > 0.0f) ? (ai / at) : 0.0f;
            if (iou > best) { best = iou; xr = xg; yr = yg; wr = wg; hr = hg; cr = cg; }
        }
        float dx = x_t - xr, dy = y_t - yr;
        float dw = sqrtf(w_t) - sqrtf(wr);
        float dh = sqrtf(h_t) - sqrtf(hr);
        loss  = LAMBDA_COORD * (dx*dx + dy*dy + dw*dw + dh*dh);
        loss += (cr - 1.0f)*(cr - 1.0f) - LAMBDA_NOOBJ * cr * cr;

        float cs = 0.0f;
#pragma unroll
        for (int j = 0; j < CLS; ++j) { float v = cell[5*NB + j]; cs += v*v; }
        float clsr = cell[5*NB + cl];
        loss += cs + (clsr - 1.0f)*(clsr - 1.0f) - clsr*clsr;
    }
    sdata[threadIdx.x] = loss;
    __syncthreads();
    for (int s = 128; s > 0; s >>= 1) {
        if (threadIdx.x < s) sdata[threadIdx.x] += sdata[threadIdx.x + s];
        __syncthreads();
    }
    if (threadIdx.x == 0) partial[blockIdx.x] = sdata[0];
}

// ---------------------------------------------------------------------------
// Kernel 2: sum of conf^2 over the whole output tensor, streamed as float4.
// The WMMA matrix pipe is used as the accumulator: A = per-lane squared
// values, B = all-ones  =>  D = rowsum(A) + C.  C/D layout: column 0 lives in
// lane 0 (M=0..7) and lane 16 (M=8..15), so wave total = those 16 f32 regs.
// ---------------------------------------------------------------------------
__global__ void conf_sq_wmma(const float* __restrict__ out,
                             float* __restrict__ partial, int n4) {
    __shared__ float sdata[256];
    v8f acc  = {0.f,0.f,0.f,0.f,0.f,0.f,0.f,0.f};
    v2f ones = {1.0f, 1.0f};

    const int stride = gridDim.x * blockDim.x;
    for (int base = blockIdx.x * blockDim.x; base < n4; base += stride) {
        int i = base + (int)threadIdx.x;          // uniform trip count per block
        float4 v = make_float4(0.f, 0.f, 0.f, 0.f);
        if (i < n4) v = ((const float4*)out)[i];  // predicated load only
        int e   = i * 4;
        int ch0 = e % CH;
        int ch1 = ch0 + 1; if (ch1 >= CH) ch1 -= CH;
        int ch2 = ch0 + 2; if (ch2 >= CH) ch2 -= CH;
        int ch3 = ch0 + 3; if (ch3 >= CH) ch3 -= CH;
        float s0 = (ch0 == 4 || ch0 == 9) ? v.x * v.x : 0.0f;
        float s1 = (ch1 == 4 || ch1 == 9) ? v.y * v.y : 0.0f;
        float s2 = (ch2 == 4 || ch2 == 9) ? v.z * v.z : 0.0f;
        float s3 = (ch3 == 4 || ch3 == 9) ? v.w * v.w : 0.0f;
        v2f a;
        a.x = s0 + s1;
        a.y = s2 + s3;
        // D = A(16x4) * ones(4x16) + C : accumulates 64 lane values per issue
        acc = __builtin_amdgcn_wmma_f32_16x16x4_f32(
                  /*neg_a=*/false, a, /*neg_b=*/false, ones,
                  /*c_mod=*/(short)0, acc, /*reuse_a=*/false, /*reuse_b=*/false);
    }
    float s = acc[0]+acc[1]+acc[2]+acc[3]+acc[4]+acc[5]+acc[6]+acc[7];
    sdata[threadIdx.x] = s;
    __syncthreads();
    if (threadIdx.x == 0) {
        float tot = 0.0f;
        // column 0 of each wave's D fragment: lane 0 (M0-7) + lane 16 (M8-15)
        for (int w = 0; w < 8; ++w) tot += sdata[w*32 + 0] + sdata[w*32 + 16];
        partial[blockIdx.x] = tot;
    }
}

// ---------------------------------------------------------------------------
// Kernel 3: deterministic final combine:  (sum_t + 0.5*sum_conf) / B_IMG
// ---------------------------------------------------------------------------
__global__ void final_reduce(const float* __restrict__ tpart, int nt,
                             const float* __restrict__ cpart, int nc,
                             float* __restrict__ outp) {
    __shared__ float sdata[256];
    float st = 0.0f, sc = 0.0f;
    for (int i = threadIdx.x; i < nt; i += 256) st += tpart[i];
    for (int i = threadIdx.x; i < nc; i += 256) sc += cpart[i];
    sdata[threadIdx.x] = st + LAMBDA_NOOBJ * sc;
    __syncthreads();
    for (int s = 128; s > 0; s >>= 1) {
        if (threadIdx.x < s) sdata[threadIdx.x] += sdata[threadIdx.x + s];
        __syncthreads();
    }
    if (threadIdx.x == 0) outp[0] = sdata[0] / (float)B_IMG;
}

extern "C" void kernel_launch(void* const* d_in, const int* in_sizes, int n_in,
                              void* d_out, int out_size, void* d_ws, size_t ws_size,
                              hipStream_t stream) {
    (void)in_sizes; (void)n_in; (void)out_size; (void)ws_size;
    const float* output = (const float*)d_in[0];
    const float* target = (const float*)d_in[1];
    float* outp = (float*)d_out;
    float* ws   = (float*)d_ws;

    const int TBLK = T_TOT / 256;                    // 512 full blocks
    const int CBLK = 2048;                           // grid-stride blocks
    const int N4   = (B_IMG * G_DIM * G_DIM * CH) / 4; // 6,021,120 float4s

    float* tpart = ws;           // [TBLK]
    float* cpart = ws + TBLK;    // [CBLK]

    yolo_target_loss<<<TBLK, 256, 0, stream>>>(output, target, tpart, T_TOT);
    conf_sq_wmma   <<<CBLK, 256, 0, stream>>>(output, cpart, N4);
    final_reduce   <<<1,    256, 0, stream>>>(tpart, TBLK, cpart, CBLK, outp);
}